// model_9251359556172
// MI455X (gfx1250) — compile-verified
//
#include <hip/hip_runtime.h>
#include <cmath>

typedef __attribute__((ext_vector_type(2))) float v2f;
typedef __attribute__((ext_vector_type(8))) float v8f;

#define NEG_SLOPE 0.2f

// Sign-aware float atomic max (lowers to native int atomic max/min in L2).
__device__ __forceinline__ void atomicMaxF(float* addr, float val) {
  if (val >= 0.0f) atomicMax((int*)addr, __float_as_int(val));
  else             atomicMin((unsigned int*)addr, __float_as_uint(val));
}

// ---------------------------------------------------------------------------
// C[M,Nc] = A[M,K] @ B[K,Nc] (+bias)(+relu), row-major, fp32 via
// V_WMMA_F32_16X16X4_F32.  Requires M%16==0, Nc%64==0, K%4==0.
// One wave computes a 16x64 output tile (4 x v8f accumulators).
// A 16x4 layout: lanes 0-15 hold {K=k,k+1} of row M=lane; lanes 16-31 hold
// {K=k+2,k+3} of row M=lane-16.  B 4x16: VGPR0 = row k (lanes 0-15) / row
// k+2 (lanes 16-31); VGPR1 = row k+1 / k+3.  D: VGPR r -> rows r / r+8.
// ---------------------------------------------------------------------------
__global__ __launch_bounds__(256) void gemm_wmma_f32(
    const float* __restrict__ A, const float* __restrict__ B,
    float* __restrict__ C, const float* __restrict__ bias,
    int M, int K, int Nc, int apply_relu)
{
  const int lane = threadIdx.x & 31;
  const int wid  = (blockIdx.x * blockDim.x + threadIdx.x) >> 5;
  const int ntiles = Nc >> 6;
  const int total  = (M >> 4) * ntiles;
  if (wid >= total) return;
  const int mt = wid / ntiles, nt = wid - mt * ntiles;
  const int m0 = mt << 4, n0 = nt << 6;
  const int half = lane >> 4, lr = lane & 15;

  v8f acc0 = {}, acc1 = {}, acc2 = {}, acc3 = {};
  const float* arow = A + (size_t)(m0 + lr) * K + 2 * half;
  for (int k = 0; k < K; k += 4) {
    v2f a;
    a.x = arow[k];
    a.y = arow[k + 1];
    const float* bp = B + (size_t)(k + 2 * half) * Nc + n0 + lr;
    v2f b0, b1, b2, b3;
    b0.x = bp[0];  b0.y = bp[Nc + 0];
    b1.x = bp[16]; b1.y = bp[Nc + 16];
    b2.x = bp[32]; b2.y = bp[Nc + 32];
    b3.x = bp[48]; b3.y = bp[Nc + 48];
    acc0 = __builtin_amdgcn_wmma_f32_16x16x4_f32(false, a, false, b0, (short)0, acc0, false, false);
    acc1 = __builtin_amdgcn_wmma_f32_16x16x4_f32(false, a, false, b1, (short)0, acc1, false, false);
    acc2 = __builtin_amdgcn_wmma_f32_16x16x4_f32(false, a, false, b2, (short)0, acc2, false, false);
    acc3 = __builtin_amdgcn_wmma_f32_16x16x4_f32(false, a, false, b3, (short)0, acc3, false, false);
  }
#pragma unroll
  for (int r = 0; r < 8; ++r) {
    const int row = m0 + r + 8 * half;
    float* crow = C + (size_t)row * Nc + n0 + lr;
    float v0 = acc0[r], v1 = acc1[r], v2 = acc2[r], v3 = acc3[r];
    if (bias) {
      v0 += bias[n0 + lr];
      v1 += bias[n0 + 16 + lr];
      v2 += bias[n0 + 32 + lr];
      v3 += bias[n0 + 48 + lr];
    }
    if (apply_relu) {
      v0 = fmaxf(v0, 0.f); v1 = fmaxf(v1, 0.f);
      v2 = fmaxf(v2, 0.f); v3 = fmaxf(v3, 0.f);
    }
    crow[0] = v0; crow[16] = v1; crow[32] = v2; crow[48] = v3;
  }
}

// ---------------------------------------------------------------------------
// Per-layer scratch init: per-(node,head) max = -inf, sum = 0, node accum = 0.
// ---------------------------------------------------------------------------
__global__ __launch_bounds__(256) void init_layer(
    float* __restrict__ nmax, float* __restrict__ nsum,
    float* __restrict__ outacc, int n)
{
  const int i = blockIdx.x * blockDim.x + threadIdx.x;
  if (i < n * 128) outacc[i] = 0.f;
  if (i < n * 4) { nmax[i] = -INFINITY; nsum[i] = 0.f; }
}

// ---------------------------------------------------------------------------
// Edge pass A (wave-per-edge): f_out = f_ni[src]+f_nj[dst]+ew@Wfij+b,
// leaky-relu, per-head attention dot -> logits; atomic max into nmax[dst].
// Lane l covers (h,f) pairs hf=l and hf=l+32 (hf = h*16+f).
// ---------------------------------------------------------------------------
__global__ __launch_bounds__(256) void edge_logits_kernel(
    const float* __restrict__ fni, const float* __restrict__ fnj,
    const float* __restrict__ ew, const int* __restrict__ src,
    const int* __restrict__ dst, const float* __restrict__ Wfij,
    const float* __restrict__ attn, const float* __restrict__ bvec,
    float* __restrict__ logits, float* __restrict__ nmax, int E)
{
  __shared__ float sW[16 * 64];
  __shared__ float sAttn[64];
  __shared__ float sB[64];
  for (int i = threadIdx.x; i < 16 * 64; i += blockDim.x) sW[i] = Wfij[i];
  if (threadIdx.x < 64) {
    sAttn[threadIdx.x] = attn[threadIdx.x];
    sB[threadIdx.x]    = bvec[threadIdx.x];
  }
  __syncthreads();

  const int lane = threadIdx.x & 31;
  const int hf0 = lane, hf1 = lane + 32;
  int wid = (blockIdx.x * blockDim.x + threadIdx.x) >> 5;
  const int nw = (gridDim.x * blockDim.x) >> 5;
  for (int e = wid; e < E; e += nw) {
    const int si = src[e], di = dst[e];
    const float ewv = ew[(size_t)e * 16 + (lane & 15)];
    float acc0 = fni[(size_t)si * 64 + hf0] + fnj[(size_t)di * 64 + hf0] + sB[hf0];
    float acc1 = fni[(size_t)si * 64 + hf1] + fnj[(size_t)di * 64 + hf1] + sB[hf1];
#pragma unroll
    for (int k = 0; k < 16; ++k) {
      const float wk = __shfl(ewv, k, 32);   // broadcast ew[e][k]
      acc0 += wk * sW[k * 64 + hf0];
      acc1 += wk * sW[k * 64 + hf1];
    }
    acc0 = (acc0 > 0.f) ? acc0 : NEG_SLOPE * acc0;
    acc1 = (acc1 > 0.f) ? acc1 : NEG_SLOPE * acc1;
    float p0 = acc0 * sAttn[hf0];
    float p1 = acc1 * sAttn[hf1];
#pragma unroll
    for (int off = 8; off >= 1; off >>= 1) {  // reduce 16 lanes (one head each)
      p0 += __shfl_xor(p0, off, 32);
      p1 += __shfl_xor(p1, off, 32);
    }
    if ((lane & 15) == 0) {
      const int h0 = lane >> 4;               // lane0 -> heads {0,2}, lane16 -> {1,3}
      logits[(size_t)e * 4 + h0]     = p0;
      logits[(size_t)e * 4 + h0 + 2] = p1;
      atomicMaxF(&nmax[(size_t)di * 4 + h0], p0);
      atomicMaxF(&nmax[(size_t)di * 4 + h0 + 2], p1);
    }
  }
}

// ---------------------------------------------------------------------------
// Edge pass B (wave-per-edge): a = exp(logit - m[dst]); nsum[dst] += a;
// outacc[dst] += ft[src]*a  (unnormalized -- divided by nsum in finalize).
// ---------------------------------------------------------------------------
__global__ __launch_bounds__(256) void edge_aggregate_kernel(
    const float* __restrict__ logits, const float* __restrict__ nmax,
    const float* __restrict__ ft, const int* __restrict__ src,
    const int* __restrict__ dst, float* __restrict__ nsum,
    float* __restrict__ outacc, int E)
{
  const int lane = threadIdx.x & 31;
  int wid = (blockIdx.x * blockDim.x + threadIdx.x) >> 5;
  const int nw = (gridDim.x * blockDim.x) >> 5;
  for (int e = wid; e < E; e += nw) {
    const int si = src[e], di = dst[e];
    const int h = lane & 3;
    const float av = expf(logits[(size_t)e * 4 + h] - nmax[(size_t)di * 4 + h]);
    if (lane < 4) unsafeAtomicAdd(&nsum[(size_t)di * 4 + lane], av);
#pragma unroll
    for (int hh = 0; hh < 4; ++hh) {
      const float ah = __shfl(av, hh, 32);
      const float v = ft[(size_t)si * 128 + hh * 32 + lane] * ah;
      unsafeAtomicAdd(&outacc[(size_t)di * 128 + hh * 32 + lane], v);
    }
  }
}

// ---------------------------------------------------------------------------
// Finalize layer in place: out = relu(acc / nsum[node,head]) (0 if no edges).
// ---------------------------------------------------------------------------
__global__ __launch_bounds__(256) void finalize_layer(
    float* __restrict__ outacc, const float* __restrict__ nsum, int n)
{
  const int i = blockIdx.x * blockDim.x + threadIdx.x;
  if (i >= n * 128) return;
  const int node = i >> 7, hf = i & 127, h = hf >> 5;
  const float sv = nsum[node * 4 + h];
  float v = (sv > 0.f) ? outacc[i] / sv : 0.f;
  outacc[i] = (v > 0.f) ? v : 0.f;
}

// ---------------------------------------------------------------------------
// Head matvec: out[n] = z[n,:] @ w + b  (wave-per-node reduction).
// ---------------------------------------------------------------------------
__global__ __launch_bounds__(256) void lin2_kernel(
    const float* __restrict__ z, const float* __restrict__ wv,
    const float* __restrict__ bv, float* __restrict__ out, int n)
{
  const int lane = threadIdx.x & 31;
  const int wid = (blockIdx.x * blockDim.x + threadIdx.x) >> 5;
  if (wid >= n) return;
  const float* zr = z + (size_t)wid * 128;
  float acc = 0.f;
#pragma unroll
  for (int k = 0; k < 4; ++k) acc += zr[lane + 32 * k] * wv[lane + 32 * k];
#pragma unroll
  for (int off = 16; off >= 1; off >>= 1) acc += __shfl_xor(acc, off, 32);
  if (lane == 0) out[wid] = acc + bv[0];
}

// ---------------------------------------------------------------------------

static inline void launch_gemm(const float* A, const float* B, float* C,
                               const float* bias, int M, int K, int Nc,
                               int relu, hipStream_t s)
{
  const int waves = (M / 16) * (Nc / 64);
  const int blocks = (waves * 32 + 255) / 256;
  gemm_wmma_f32<<<blocks, 256, 0, s>>>(A, B, C, bias, M, K, Nc, relu);
}

extern "C" void kernel_launch(void* const* d_in, const int* in_sizes, int n_in,
                              void* d_out, int out_size, void* d_ws, size_t ws_size,
                              hipStream_t stream)
{
  const int N = in_sizes[0] / 128;   // 50000
  const int E = in_sizes[2];         // 800000

  const float* x   = (const float*)d_in[0];
  const float* ew  = (const float*)d_in[1];
  const int*   src = (const int*)d_in[2];
  const int*   dst = (const int*)d_in[3];
  const float *Wn[3], *Wni[3], *Wnj[3], *Wfij[3], *attn[3], *bvec[3];
  for (int l = 0; l < 3; ++l) {
    const int base = 4 + 6 * l;
    Wn[l]   = (const float*)d_in[base + 0];
    Wni[l]  = (const float*)d_in[base + 1];
    Wnj[l]  = (const float*)d_in[base + 2];
    Wfij[l] = (const float*)d_in[base + 3];
    attn[l] = (const float*)d_in[base + 4];
    bvec[l] = (const float*)d_in[base + 5];
  }
  const float* lin1_w = (const float*)d_in[22];
  const float* lin1_b = (const float*)d_in[23];
  const float* lin2_w = (const float*)d_in[24];
  const float* lin2_b = (const float*)d_in[25];

  // Workspace carve-up (~117 MB of fp32).
  float* w = (float*)d_ws;
  float* bufA = w; w += (size_t)N * 128;   // layer output ping
  float* bufB = w; w += (size_t)N * 128;   // layer output pong
  float* ftb  = w; w += (size_t)N * 128;   // ft = x@Wn (reused as lin1 output)
  float* fni  = w; w += (size_t)N * 64;
  float* fnj  = w; w += (size_t)N * 64;
  float* lgt  = w; w += (size_t)E * 4;
  float* nmax = w; w += (size_t)N * 4;
  float* nsum = w; w += (size_t)N * 4;

  const int eblocks    = (E + 7) / 8;          // 8 edge-waves per 256-thread block
  const int nblocks128 = (N * 128 + 255) / 256;

  const float* hin = x;
  float* houts[3] = { bufA, bufB, bufA };
  for (int l = 0; l < 3; ++l) {
    float* hout = houts[l];
    launch_gemm(hin, Wni[l], fni, nullptr, N, 128, 64, 0, stream);
    launch_gemm(hin, Wnj[l], fnj, nullptr, N, 128, 64, 0, stream);
    launch_gemm(hin, Wn[l],  ftb, nullptr, N, 128, 128, 0, stream);
    init_layer<<<nblocks128, 256, 0, stream>>>(nmax, nsum, hout, N);
    edge_logits_kernel<<<eblocks, 256, 0, stream>>>(
        fni, fnj, ew, src, dst, Wfij[l], attn[l], bvec[l], lgt, nmax, E);
    edge_aggregate_kernel<<<eblocks, 256, 0, stream>>>(
        lgt, nmax, ftb, src, dst, nsum, hout, E);
    finalize_layer<<<nblocks128, 256, 0, stream>>>(hout, nsum, N);
    hin = hout;
  }

  // Head: relu(h @ lin1_w + lin1_b) @ lin2_w + lin2_b
  launch_gemm(hin, lin1_w, ftb, lin1_b, N, 128, 128, 1, stream);
  {
    const int blocks = (N * 32 + 255) / 256;
    lin2_kernel<<<blocks, 256, 0, stream>>>(ftb, lin2_w, lin2_b, (float*)d_out, N);
  }
}